// IFSpike1_18923625906304
// MI455X (gfx1250) — compile-verified
//
#include <hip/hip_runtime.h>
#include <stdint.h>

// IF-neuron scan: out[t] = heaviside(mem + x[t] - 1), mem soft-reset.
// T = 8 time planes, S spatial elements per plane. Pure HBM-streaming kernel
// (512 MB total traffic, ~22us floor at 23.3 TB/s):
//   - stage inputs with CDNA5 GLOBAL_LOAD_ASYNC_TO_LDS_B128 (ASYNCcnt),
//     non-temporal since each input byte is read exactly once
//   - pipelined s_wait_asynccnt (loads complete in order) so plane-t compute
//     and stores overlap with planes t+1..7 still in flight
//   - non-temporal 128-bit stores (output written once, never re-read)

typedef float v4f __attribute__((ext_vector_type(4)));

#define T_STEPS 8
#define VEC     4
#define BLOCK   256

__device__ __forceinline__ void async_load_b128_to_lds(uint32_t lds_byte_addr,
                                                       const float* gptr) {
    // GLOBAL_LOAD_ASYNC_TO_LDS_B128, GV mode (64-bit vaddr, no saddr).
    // VDST = LDS byte address VGPR, VADDR = 64-bit global address VGPR pair.
    // th:TH_LOAD_NT: streaming input (256MB > 192MB L2), read exactly once.
    asm volatile("global_load_async_to_lds_b128 %0, %1, off th:TH_LOAD_NT"
                 :
                 : "v"(lds_byte_addr), "v"((uint64_t)(uintptr_t)gptr)
                 : "memory");
}

#if __has_builtin(__builtin_amdgcn_s_wait_asynccnt)
#define WAIT_ASYNC(n) __builtin_amdgcn_s_wait_asynccnt(n)
#else
#define WAIT_ASYNC(n) asm volatile("s_wait_asynccnt %0" ::"i"(n) : "memory")
#endif

__global__ __launch_bounds__(BLOCK) void if_spike_kernel(
        const float* __restrict__ x, float* __restrict__ out, long long S) {
    __shared__ v4f tile[T_STEPS][BLOCK];   // 8 * 256 * 16B = 32 KB per block

    const int tid = threadIdx.x;
    const long long sb = ((long long)blockIdx.x * BLOCK + tid) * VEC;
    if (sb >= S) return;

    // Issue all 8 async loads back-to-back: 8 in-flight B128 transfers per
    // lane, tracked by ASYNCcnt. Planes are 32MB apart; within a plane the
    // wave's 32 lanes cover 512 contiguous bytes (fully coalesced).
#pragma unroll
    for (int t = 0; t < T_STEPS; ++t) {
        const float* g = x + (long long)t * S + sb;
        uint32_t laddr = (uint32_t)(uintptr_t)&tile[t][tid];  // LDS offset = low 32 bits
        async_load_b128_to_lds(laddr, g);
    }

    v4f m = {0.5f, 0.5f, 0.5f, 0.5f};

    // Each lane consumes only its own LDS slots -> per-wave ASYNCcnt waits are
    // sufficient, no workgroup barrier required. Async loads complete in
    // order, so plane t is in LDS once ASYNCcnt <= 7-t.
    auto step = [&](int t) {
        v4f v = tile[t][tid];      // ds_load_b128
        v4f s;
        m.x += v.x; s.x = (m.x >= 1.0f) ? 1.0f : 0.0f; m.x -= s.x;
        m.y += v.y; s.y = (m.y >= 1.0f) ? 1.0f : 0.0f; m.y -= s.y;
        m.z += v.z; s.z = (m.z >= 1.0f) ? 1.0f : 0.0f; m.z -= s.z;
        m.w += v.w; s.w = (m.w >= 1.0f) ? 1.0f : 0.0f; m.w -= s.w;
        // Streaming output, never re-read: non-temporal 128-bit store.
        __builtin_nontemporal_store(s, (v4f*)(out + (long long)t * S + sb));
    };

    WAIT_ASYNC(7); step(0);
    WAIT_ASYNC(6); step(1);
    WAIT_ASYNC(5); step(2);
    WAIT_ASYNC(4); step(3);
    WAIT_ASYNC(3); step(4);
    WAIT_ASYNC(2); step(5);
    WAIT_ASYNC(1); step(6);
    WAIT_ASYNC(0); step(7);
}

extern "C" void kernel_launch(void* const* d_in, const int* in_sizes, int n_in,
                              void* d_out, int out_size, void* d_ws, size_t ws_size,
                              hipStream_t stream) {
    (void)n_in; (void)d_ws; (void)ws_size; (void)out_size;
    const float* x = (const float*)d_in[0];
    float* out = (float*)d_out;

    const long long total = (long long)in_sizes[0];   // T * S
    const long long S = total / T_STEPS;              // 8,388,608 spatial elems
    const long long threads_needed = (S + VEC - 1) / VEC;
    const int nblocks = (int)((threads_needed + BLOCK - 1) / BLOCK);  // 8192

    if_spike_kernel<<<nblocks, BLOCK, 0, stream>>>(x, out, S);
}